// Winograd_69088843923578
// MI455X (gfx1250) — compile-verified
//
#include <hip/hip_runtime.h>

typedef __attribute__((ext_vector_type(2))) float v2f;
typedef __attribute__((ext_vector_type(8))) float v8f;

// Problem constants (fixed by the reference harness)
#define NB   16      // batch
#define CC   128     // in channels
#define KK   128     // out channels
#define HH   56
#define WW   56
#define TH   28      // tiles per dim (F(2x2,3x3), stride 2 over padded 58)
#define PP   (NB*TH*TH)   // 12544 tiles total

// ---------------------------------------------------------------------------
// Filter transform: U[t][k][c] = (G g G^T)[i][j], t = i*4+j
// U stored row-major over c => fp32 WMMA A-fragment is one b64 load.
// ---------------------------------------------------------------------------
__global__ void wino_filter_xform(const float* __restrict__ g,
                                  float* __restrict__ U) {
    int idx = blockIdx.x * blockDim.x + threadIdx.x;
    if (idx >= KK * CC) return;
    int c = idx % CC;
    int k = idx / CC;

    const float* gp = g + ((size_t)k * CC + c) * 9;
    float w[3][3];
#pragma unroll
    for (int i = 0; i < 3; ++i)
#pragma unroll
        for (int j = 0; j < 3; ++j) w[i][j] = gp[i * 3 + j];

    float tmp[4][3];
#pragma unroll
    for (int j = 0; j < 3; ++j) {
        tmp[0][j] = w[0][j];
        tmp[1][j] = 0.5f * (w[0][j] + w[1][j] + w[2][j]);
        tmp[2][j] = 0.5f * (w[0][j] - w[1][j] + w[2][j]);
        tmp[3][j] = w[2][j];
    }
#pragma unroll
    for (int i = 0; i < 4; ++i) {
        float u0 = tmp[i][0];
        float u1 = 0.5f * (tmp[i][0] + tmp[i][1] + tmp[i][2]);
        float u2 = 0.5f * (tmp[i][0] - tmp[i][1] + tmp[i][2]);
        float u3 = tmp[i][2];
        float uu[4] = {u0, u1, u2, u3};
#pragma unroll
        for (int j = 0; j < 4; ++j) {
            int t = i * 4 + j;
            U[((size_t)t * KK + k) * CC + c] = uu[j];
        }
    }
}

// ---------------------------------------------------------------------------
// Input transform: V[t][c][p] = (B^T d B)[i][j], stored c-pair interleaved:
//   float index = ((t*64 + c/2)*P + p)*2 + (c&1)
// so the fp32 WMMA B-fragment (K=2*half..2*half+1, N=lane) is one float2.
// ---------------------------------------------------------------------------
__global__ void wino_input_xform(const float* __restrict__ x,
                                 float* __restrict__ V) {
    int idx = blockIdx.x * blockDim.x + threadIdx.x;
    if (idx >= CC * PP) return;
    int p = idx % PP;
    int c = idx / PP;
    int tw = p % TH;
    int th = (p / TH) % TH;
    int n  = p / (TH * TH);

    const float* xp = x + ((size_t)n * CC + c) * (HH * WW);
    float d[4][4];
#pragma unroll
    for (int i = 0; i < 4; ++i) {
        int r = 2 * th + i - 1;
#pragma unroll
        for (int j = 0; j < 4; ++j) {
            int q = 2 * tw + j - 1;
            bool ok = (r >= 0) & (r < HH) & (q >= 0) & (q < WW);
            d[i][j] = ok ? xp[r * WW + q] : 0.0f;
        }
    }
    float tm[4][4];
#pragma unroll
    for (int j = 0; j < 4; ++j) {
        tm[0][j] = d[0][j] - d[2][j];
        tm[1][j] = d[1][j] + d[2][j];
        tm[2][j] = d[2][j] - d[1][j];
        tm[3][j] = d[1][j] - d[3][j];
    }
#pragma unroll
    for (int i = 0; i < 4; ++i) {
        float v0 = tm[i][0] - tm[i][2];
        float v1 = tm[i][1] + tm[i][2];
        float v2 = tm[i][2] - tm[i][1];
        float v3 = tm[i][1] - tm[i][3];
        float vv[4] = {v0, v1, v2, v3};
#pragma unroll
        for (int j = 0; j < 4; ++j) {
            int t = i * 4 + j;
            size_t a = (((size_t)t * (CC / 2) + (c >> 1)) * PP + p) * 2 + (c & 1);
            V[a] = vv[j];
        }
    }
}

// ---------------------------------------------------------------------------
// Batched Winograd-domain GEMM: M[t] = U[t](128x128) * V[t](128xP)
// Block = 128 threads = 4 waves; wave w computes M rows [32w, 32w+32) for a
// 64-wide P slice. The shared B slice (V) is staged into LDS with
// GLOBAL_LOAD_ASYNC_TO_LDS_B128, double-buffered in 2KB chunks of K=8,
// waited with s_wait_asynccnt; B fragments come from ds_load_b64.
// Grid: (P/64, 1, 16).
// ---------------------------------------------------------------------------
#define WMMA_F32(ACC, A, B) \
    ACC = __builtin_amdgcn_wmma_f32_16x16x4_f32(false, (A), false, (B), \
                                                (short)0, (ACC), false, false)

__global__ void __launch_bounds__(128)
wino_gemm(const float* __restrict__ U,   // [16][128][128]
          const float* __restrict__ V,   // pair-interleaved [16][64][P][2]
          float* __restrict__ M) {       // [16][128][P]
    // double-buffered B stage: [buf][pair-row(4)][col(64)][2 floats] = 2KB/buf
    __shared__ float sB[2][4 * 64 * 2];

    const int t    = blockIdx.z;
    const int n0   = blockIdx.x * 64;
    const int wave = threadIdx.x >> 5;
    const int lane = threadIdx.x & 31;
    const int half = lane >> 4;          // selects K pair {0,1} vs {2,3}
    const int l16  = lane & 15;
    const int m0   = wave * 32;

    // A row pointers for this lane (rows m0+l16 and m0+16+l16)
    const float* Ur0 = U + ((size_t)t * KK + (m0 + l16)) * CC;
    const float* Ur1 = Ur0 + 16 * CC;

    // global base of this block's V slice (t, col n0), bytes via lane offsets
    const float* vslice = V + (size_t)t * (CC / 2) * (size_t)PP * 2 + (size_t)n0 * 2;
    const unsigned lds0 = (unsigned)(size_t)(&sB[0][0]);

    // chunk ch covers pair-rows 4ch..4ch+3 (c = 8ch..8ch+7); wave w copies
    // pair-row 4ch+w (512B), one b128 per lane.
    const int NCH = CC / 8;   // 16 chunks

    v8f a00 = {}, a01 = {}, a02 = {}, a03 = {};   // rows m0..m0+15
    v8f a10 = {}, a11 = {}, a12 = {}, a13 = {};   // rows m0+16..m0+31

    // prologue: issue chunk 0 into buffer 0
    {
        unsigned long long ga = (unsigned long long)(size_t)vslice
                              + (unsigned long long)(0 * 4 + wave) * (PP * 8ull)
                              + (unsigned long long)lane * 16;
        unsigned la = lds0 + 0 * 2048 + wave * 512 + lane * 16;
        asm volatile("global_load_async_to_lds_b128 %0, %1, off"
                     :: "v"(la), "v"(ga) : "memory");
    }

    for (int ch = 0; ch < NCH; ++ch) {
        const int buf = ch & 1;
        if (ch + 1 < NCH) {
            unsigned long long ga = (unsigned long long)(size_t)vslice
                                  + (unsigned long long)((ch + 1) * 4 + wave) * (PP * 8ull)
                                  + (unsigned long long)lane * 16;
            unsigned la = lds0 + (buf ^ 1) * 2048 + wave * 512 + lane * 16;
            asm volatile("global_load_async_to_lds_b128 %0, %1, off"
                         :: "v"(la), "v"(ga) : "memory");
            asm volatile("s_wait_asynccnt 0x1" ::: "memory");  // chunk ch landed
        } else {
            asm volatile("s_wait_asynccnt 0x0" ::: "memory");
        }
        __syncthreads();   // chunk ch visible to all waves

        const float* sb = &sB[buf][0];
#pragma unroll
        for (int s = 0; s < 2; ++s) {          // two K=4 steps per chunk
            const int c0 = ch * 8 + s * 4;
            v2f aA = *(const v2f*)(Ur0 + c0 + 2 * half);
            v2f aB = *(const v2f*)(Ur1 + c0 + 2 * half);

            const float* srow = sb + (size_t)(s * 2 + half) * 64 * 2;
            v2f b0 = *(const v2f*)(srow + 2 * (0 * 16 + l16));
            v2f b1 = *(const v2f*)(srow + 2 * (1 * 16 + l16));
            v2f b2 = *(const v2f*)(srow + 2 * (2 * 16 + l16));
            v2f b3 = *(const v2f*)(srow + 2 * (3 * 16 + l16));

            WMMA_F32(a00, aA, b0);
            WMMA_F32(a01, aA, b1);
            WMMA_F32(a02, aA, b2);
            WMMA_F32(a03, aA, b3);
            WMMA_F32(a10, aB, b0);
            WMMA_F32(a11, aB, b1);
            WMMA_F32(a12, aB, b2);
            WMMA_F32(a13, aB, b3);
        }
        __syncthreads();   // everyone done reading buf before it is reissued
    }

    // D layout: VGPR v -> row (m-base + v + 8*half), col n0 + 16*nt + l16
    float* Mt = M + ((size_t)t * KK + m0) * (size_t)PP;
#pragma unroll
    for (int v = 0; v < 8; ++v) {
        size_t r0 = (size_t)(v + 8 * half) * PP;
        size_t r1 = r0 + (size_t)16 * PP;
        Mt[r0 + n0 + 0 * 16 + l16] = a00[v];
        Mt[r0 + n0 + 1 * 16 + l16] = a01[v];
        Mt[r0 + n0 + 2 * 16 + l16] = a02[v];
        Mt[r0 + n0 + 3 * 16 + l16] = a03[v];
        Mt[r1 + n0 + 0 * 16 + l16] = a10[v];
        Mt[r1 + n0 + 1 * 16 + l16] = a11[v];
        Mt[r1 + n0 + 2 * 16 + l16] = a12[v];
        Mt[r1 + n0 + 3 * 16 + l16] = a13[v];
    }
}

// ---------------------------------------------------------------------------
// Inverse transform: Y(2x2) = A^T M A per (k, tile); 16 coalesced loads from
// M[t][k][p], float2 stores into NCHW output.
// ---------------------------------------------------------------------------
__global__ void wino_output_xform(const float* __restrict__ M,
                                  float* __restrict__ Y) {
    int idx = blockIdx.x * blockDim.x + threadIdx.x;
    if (idx >= KK * PP) return;
    int p = idx % PP;
    int k = idx / PP;
    int tw = p % TH;
    int th = (p / TH) % TH;
    int n  = p / (TH * TH);

    float m[4][4];
#pragma unroll
    for (int i = 0; i < 4; ++i)
#pragma unroll
        for (int j = 0; j < 4; ++j) {
            int t = i * 4 + j;
            m[i][j] = M[((size_t)t * KK + k) * (size_t)PP + p];
        }

    float r0[4], r1[4];
#pragma unroll
    for (int j = 0; j < 4; ++j) {
        r0[j] = m[0][j] + m[1][j] + m[2][j];
        r1[j] = m[1][j] - m[2][j] - m[3][j];
    }
    float y00 = r0[0] + r0[1] + r0[2];
    float y01 = r0[1] - r0[2] - r0[3];
    float y10 = r1[0] + r1[1] + r1[2];
    float y11 = r1[1] - r1[2] - r1[3];

    float* yp = Y + (((size_t)n * KK + k) * HH + 2 * th) * WW + 2 * tw;
    *(v2f*)(yp)      = (v2f){y00, y01};
    *(v2f*)(yp + WW) = (v2f){y10, y11};
}

// ---------------------------------------------------------------------------
extern "C" void kernel_launch(void* const* d_in, const int* in_sizes, int n_in,
                              void* d_out, int out_size, void* d_ws, size_t ws_size,
                              hipStream_t stream) {
    (void)in_sizes; (void)n_in; (void)out_size; (void)ws_size;
    const float* x = (const float*)d_in[0];   // (16,128,56,56) fp32
    const float* g = (const float*)d_in[1];   // (128,128,3,3)  fp32
    float* out = (float*)d_out;               // (16,128,56,56) fp32

    float* U = (float*)d_ws;                                   // 1 MB
    float* V = U + (size_t)16 * KK * CC;                       // 103 MB
    float* M = V + (size_t)16 * (CC / 2) * (size_t)PP * 2;     // 103 MB

    wino_filter_xform<<<(KK * CC + 255) / 256, 256, 0, stream>>>(g, U);
    wino_input_xform<<<((size_t)CC * PP + 255) / 256, 256, 0, stream>>>(x, V);

    dim3 gg(PP / 64, 1, 16);   // 196 x 1 x 16 blocks
    wino_gemm<<<gg, 128, 0, stream>>>(U, V, M);

    wino_output_xform<<<((size_t)KK * PP + 255) / 256, 256, 0, stream>>>(M, out);
}